// TripletLoss_39058432590300
// MI455X (gfx1250) — compile-verified
//
#include <hip/hip_runtime.h>
#include <hip/hip_bf16.h>

typedef __attribute__((ext_vector_type(2))) float v2f;
typedef __attribute__((ext_vector_type(8))) float v8f;

#define BATCH 384
#define DIM   256
#define MARGIN_F 0.2f

// Workspace layout (in floats):
//   [0, 384)                      : r[i]     = ||x_i||^2
//   [384, 768)                    : counts[i] = class size of label[i] (as float)
//   [768, 768 + 384*384)          : dist2[i*384 + j]
//   [768 + 384*384, +2)           : accum[0]=num, accum[1]=den
#define OFF_R    0
#define OFF_CNT  384
#define OFF_D2   768
#define OFF_ACC  (768 + BATCH * BATCH)

// ---------------------------------------------------------------------------
// Kernel 1: row norms + class counts + zero accumulators.  <<<1, 384>>>
// ---------------------------------------------------------------------------
__global__ void prep_kernel(const float* __restrict__ x,
                            const int* __restrict__ labels,
                            float* __restrict__ ws) {
    int i = threadIdx.x;              // 0..383
    if (i < 2) ws[OFF_ACC + i] = 0.0f;
    if (i >= BATCH) return;

    // ||x_i||^2 with float4 loads (rows are 1KB-aligned)
    const float4* row = (const float4*)(x + (size_t)i * DIM);
    float s = 0.0f;
    #pragma unroll 4
    for (int k = 0; k < DIM / 4; ++k) {
        float4 v = row[k];
        s += v.x * v.x + v.y * v.y + v.z * v.z + v.w * v.w;
    }
    ws[OFF_R + i] = s;

    int la = labels[i];
    int c = 0;
    for (int j = 0; j < BATCH; ++j) c += (labels[j] == la) ? 1 : 0;
    ws[OFF_CNT + i] = (float)c;
}

// ---------------------------------------------------------------------------
// Kernel 2: dist2 tiles via f32 WMMA.  One wave per 16x16 tile, grid (24,24).
//   G = X_i * X_j^T accumulated over K=256 in steps of 4 using
//   v_wmma_f32_16x16x4_f32, then dist2 = r_a + r_p - 2*G.
//
// Fragment layouts (CDNA5 ISA 7.12.2, wave32):
//   A 16x4 (MxK):  lane L, VGPR v -> A[M = L%16][K = 2*(L/16) + v]
//   B 4x16 (KxN):  lane L, VGPR v -> B[K = 2*(L/16) + v][N = L%16]
//   Since B = X_j^T, B[k][n] = x[j_base+n][k]: identical per-lane float2 load.
//   C/D 16x16:     lane L, VGPR v -> D[M = v + 8*(L/16)][N = L%16]
// ---------------------------------------------------------------------------
__global__ __launch_bounds__(32) void gram_dist2_kernel(
        const float* __restrict__ x,
        const float* __restrict__ ws_r,
        float* __restrict__ dist2) {
    const int ti   = blockIdx.x;          // row-tile  (0..23)
    const int tj   = blockIdx.y;          // col-tile  (0..23)
    const int lane = threadIdx.x;         // 0..31, whole wave active (EXEC all-1s)
    const int m    = lane & 15;
    const int kh   = (lane >> 4) << 1;    // 0 or 2: K-pair selected by lane half

    const float* arow = x + (size_t)(ti * 16 + m) * DIM + kh;
    const float* brow = x + (size_t)(tj * 16 + m) * DIM + kh;

    v8f c = {};
    #pragma unroll 8
    for (int k = 0; k < DIM; k += 4) {
        v2f a = *(const v2f*)(arow + k);   // 8B aligned: (k + kh) is even
        v2f b = *(const v2f*)(brow + k);
        // 8 args: (neg_a, A, neg_b, B, c_mod, C, reuse_a, reuse_b)
        c = __builtin_amdgcn_wmma_f32_16x16x4_f32(
                false, a, false, b, (short)0, c, false, false);
    }

    // Epilogue: dist2[row][col] = r[row] + r[col] - 2*G
    const int col   = tj * 16 + m;
    const int rbase = ti * 16 + ((lane >> 4) << 3);
    const float rc  = ws_r[col];
    #pragma unroll
    for (int v = 0; v < 8; ++v) {
        const int row = rbase + v;
        dist2[(size_t)row * BATCH + col] = ws_r[row] + rc - 2.0f * c[v];
    }
}

// ---------------------------------------------------------------------------
// Kernel 3: masked triple reduction. One thread per (a,p) pair.
//   <<<576, 256>>>  (576*256 == 384*384)
// ---------------------------------------------------------------------------
__global__ __launch_bounds__(256) void triplet_reduce_kernel(
        const float* __restrict__ dist2,
        const int*   __restrict__ labels,
        const float* __restrict__ counts,
        float* __restrict__ accum) {
    __shared__ int   slab[BATCH];
    __shared__ float snum[256];
    __shared__ float sden[256];

    for (int j = threadIdx.x; j < BATCH; j += blockDim.x) slab[j] = labels[j];
    __syncthreads();

    const int t = blockIdx.x * 256 + threadIdx.x;   // pair id, < 384*384
    const int a = t / BATCH;
    const int p = t - a * BATCH;

    float num = 0.0f, den = 0.0f;
    const int la = slab[a];
    if (a < p && slab[p] == la) {
        const float w   = counts[a];
        const float* dr = dist2 + (size_t)a * BATCH;
        const float thr = MARGIN_F + dr[p];         // margin + d(a,p)
        for (int n = 0; n < BATCH; ++n) {
            if (slab[n] != la) {
                num += fmaxf(thr - dr[n], 0.0f);
                den += 1.0f;
            }
        }
        num *= w;
        den *= w;
    }

    snum[threadIdx.x] = num;
    sden[threadIdx.x] = den;
    __syncthreads();
    for (int s = 128; s > 0; s >>= 1) {
        if (threadIdx.x < s) {
            snum[threadIdx.x] += snum[threadIdx.x + s];
            sden[threadIdx.x] += sden[threadIdx.x + s];
        }
        __syncthreads();
    }
    if (threadIdx.x == 0) {
        atomicAdd(&accum[0], snum[0]);
        atomicAdd(&accum[1], sden[0]);
    }
}

// ---------------------------------------------------------------------------
// Kernel 4: finalize scalar output.
// ---------------------------------------------------------------------------
__global__ void finalize_kernel(const float* __restrict__ accum,
                                float* __restrict__ out) {
    if (threadIdx.x == 0 && blockIdx.x == 0) {
        const float num = accum[0];
        const float den = accum[1];
        out[0] = (den > 0.0f) ? (num / den) : 0.0f;
    }
}

// ---------------------------------------------------------------------------
extern "C" void kernel_launch(void* const* d_in, const int* in_sizes, int n_in,
                              void* d_out, int out_size, void* d_ws, size_t ws_size,
                              hipStream_t stream) {
    const float* x      = (const float*)d_in[0];   // [384, 256] f32
    const int*   labels = (const int*)d_in[1];     // [384] i32
    float*       out    = (float*)d_out;           // scalar f32
    float*       ws     = (float*)d_ws;

    (void)in_sizes; (void)n_in; (void)out_size; (void)ws_size;

    // 1) norms + counts + zero accumulators
    prep_kernel<<<1, 384, 0, stream>>>(x, labels, ws);

    // 2) pairwise squared distances via f32 WMMA (24x24 tiles, 1 wave each)
    dim3 tiles(BATCH / 16, BATCH / 16);
    gram_dist2_kernel<<<tiles, 32, 0, stream>>>(x, ws + OFF_R, ws + OFF_D2);

    // 3) weighted batch-all triplet reduction
    triplet_reduce_kernel<<<(BATCH * BATCH) / 256, 256, 0, stream>>>(
        ws + OFF_D2, labels, ws + OFF_CNT, ws + OFF_ACC);

    // 4) finalize
    finalize_kernel<<<1, 32, 0, stream>>>(ws + OFF_ACC, out);
}